// NCC_77352361001074
// MI455X (gfx1250) — compile-verified
//
#include <hip/hip_runtime.h>

typedef __attribute__((ext_vector_type(2))) float v2f;
typedef __attribute__((ext_vector_type(8))) float v8f;
typedef __attribute__((ext_vector_type(4))) int   v4i;

#define BATCH 2
#define DD 160
#define HH 192
#define WW 224
#define ZC  16                                   // z-chunk per block (160/10)
#define NZC (DD / ZC)                            // 10
#define NBLK ((WW/16)*(HH/16)*(BATCH*NZC))       // 14*12*20 = 3360

__device__ __forceinline__ v8f wmma4(v2f a, v2f b, v8f c) {
  // D = A(16x4,f32) * B(4x16,f32) + C(16x16,f32)
  return __builtin_amdgcn_wmma_f32_16x16x4_f32(
      /*neg_a=*/false, a, /*neg_b=*/false, b,
      /*c_mod=*/(short)0, c, /*reuse_a=*/false, /*reuse_b=*/false);
}

#if __has_builtin(__builtin_amdgcn_global_load_async_to_lds_b128)
#define HAVE_ASYNC_LDS 1
typedef __attribute__((address_space(1))) v4i GV4;   // global-AS vector of 4 dwords
typedef __attribute__((address_space(3))) v4i LV4;   // LDS-AS vector of 4 dwords
#else
#define HAVE_ASYNC_LDS 0
#endif

__global__ __launch_bounds__(32)
void ncc_main(const float* __restrict__ I, const float* __restrict__ J,
              float* __restrict__ partials) {
  __shared__ float ldsIn[2][22][32];      // staged I/J halo tile (5.5 KB)
  __shared__ float tmpW[5][24][16];       // W-pass result, 22 rows + 2 zero pad
  __shared__ float ring[5][7][16][16];    // 7-deep z ring of WH-summed planes (35 KB)
  __shared__ float red[32];

  const int lane = threadIdx.x;   // 0..31
  const int half = lane >> 4;     // lane-half selects K pair in A/B operands
  const int mn   = lane & 15;     // row (A / C-D M) or col (B / C-D N) index
  const int w0 = blockIdx.x * 16;
  const int h0 = blockIdx.y * 16;
  const int b  = blockIdx.z / NZC;
  const int z0 = (blockIdx.z % NZC) * ZC;

  // block-uniform: is the full halo tile (rows h0-3..h0+18, cols w0-4..w0+27) in bounds?
  const bool inter = (w0 >= 4) && (w0 + 28 <= WW) && (h0 >= 3) && (h0 + 19 <= HH);

  // zero the two pad rows of tmpW (never written by step 1)
  for (int f = 0; f < 5; ++f) tmpW[f][22 + half][mn] = 0.0f;
  __syncthreads();

  float csum = 0.0f;

  for (int t = 0; t < ZC + 6; ++t) {
    const int zp   = z0 - 3 + t;     // input plane being WH-summed
    const int slot = t % 7;

    if (zp < 0 || zp >= DD) {        // zero-padded plane (uniform branch)
      for (int f = 0; f < 5; ++f)
        #pragma unroll
        for (int r = 0; r < 8; ++r)
          ring[f][slot][r + 8*half][mn] = 0.0f;
      __syncthreads();
    } else {
      const size_t planeBase = ((size_t)(b * DD + zp)) * HH * WW;

      // ---- Stage 22x32 halo rows of I and J into LDS, coalesced ----
      // 2 fields * 22 rows * 8 quads = 352 float4 tasks, 11 per lane
      if (inter) {
        #pragma unroll
        for (int it = 0; it < 11; ++it) {
          const int task = it * 32 + lane;
          const int f   = task / 176;
          const int rq  = task % 176;
          const int row = rq >> 3;
          const int q   = rq & 7;
          const float* src = (f ? J : I) + planeBase
                           + (size_t)(h0 - 3 + row) * WW + (w0 - 4) + 4 * q;
          float* dst = &ldsIn[f][row][4 * q];
#if HAVE_ASYNC_LDS
          __builtin_amdgcn_global_load_async_to_lds_b128(
              (GV4*)src, (LV4*)dst, /*offset=*/0, /*cpol=*/0);
#else
          *(float4*)dst = *(const float4*)src;
#endif
        }
#if HAVE_ASYNC_LDS
        asm volatile("s_wait_asynccnt 0x0" ::: "memory");
#endif
        __syncthreads();
      } else {
        // border blocks: element-masked scalar staging (zero padding)
        #pragma unroll
        for (int it = 0; it < 11; ++it) {
          const int task = it * 32 + lane;
          const int f   = task / 176;
          const int rq  = task % 176;
          const int row = rq >> 3;
          const int q   = rq & 7;
          const int hh  = h0 - 3 + row;
          const float* src = f ? J : I;
          #pragma unroll
          for (int e = 0; e < 4; ++e) {
            const int ww = w0 - 4 + 4 * q + e;
            float v = 0.0f;
            if (hh >= 0 && hh < HH && ww >= 0 && ww < WW)
              v = src[planeBase + (size_t)hh * WW + ww];
            ldsIn[f][row][4 * q + e] = v;
          }
        }
        __syncthreads();
      }

      // ---- Step 1: W-direction 7-tap sums, two 16-row groups (22 rows) ----
      // data[m][k] = input at w = w0-3+k  ->  ldsIn col k+1
      for (int g = 0; g < 2; ++g) {
        v8f cI = {}, cJ = {}, cII = {}, cJJ = {}, cIJ = {};
        const int rowInTile = 16*g + mn;         // A-operand row m for this lane
        const bool rowOK = (rowInTile < 22);
        #pragma unroll
        for (int c = 0; c < 6; ++c) {
          const int kk = 4*c + 2*half;           // K index pair base
          float fI0 = 0.f, fI1 = 0.f, fJ0 = 0.f, fJ1 = 0.f;
          if (rowOK) {
            fI0 = ldsIn[0][rowInTile][kk + 1];
            fI1 = ldsIn[0][rowInTile][kk + 2];
            fJ0 = ldsIn[1][rowInTile][kk + 1];
            fJ1 = ldsIn[1][rowInTile][kk + 2];
          }
          v2f aI  = { fI0, fI1 };
          v2f aJ  = { fJ0, fJ1 };
          v2f aII = { fI0*fI0, fI1*fI1 };
          v2f aJJ = { fJ0*fJ0, fJ1*fJ1 };
          v2f aIJ = { fI0*fJ0, fI1*fJ1 };
          // BandW[k][n] = 1 iff n <= k <= n+6   (lane's B col n = mn)
          v2f bw = { (mn <= kk   && kk   <= mn+6) ? 1.0f : 0.0f,
                     (mn <= kk+1 && kk+1 <= mn+6) ? 1.0f : 0.0f };
          cI  = wmma4(aI,  bw, cI);
          cJ  = wmma4(aJ,  bw, cJ);
          cII = wmma4(aII, bw, cII);
          cJJ = wmma4(aJJ, bw, cJJ);
          cIJ = wmma4(aIJ, bw, cIJ);
        }
        #pragma unroll
        for (int r = 0; r < 8; ++r) {
          const int row = 16*g + r + 8*half;     // C/D layout: M = r + 8*half
          if (row < 22) {
            tmpW[0][row][mn] = cI[r];
            tmpW[1][row][mn] = cJ[r];
            tmpW[2][row][mn] = cII[r];
            tmpW[3][row][mn] = cJJ[r];
            tmpW[4][row][mn] = cIJ[r];
          }
        }
      }
      __syncthreads();

      // ---- Step 2: H-direction 7-tap sums: BandH(16x24) x tmpW(24x16) ----
      v8f sI = {}, sJ = {}, sII = {}, sJJ = {}, sIJ = {};
      #pragma unroll
      for (int c = 0; c < 6; ++c) {
        const int kk = 4*c + 2*half;
        // BandH[m][k] = 1 iff m <= k <= m+6    (lane's A row m = mn)
        v2f ah = { (mn <= kk   && kk   <= mn+6) ? 1.0f : 0.0f,
                   (mn <= kk+1 && kk+1 <= mn+6) ? 1.0f : 0.0f };
        v2f bI  = { tmpW[0][kk][mn], tmpW[0][kk+1][mn] };
        v2f bJ  = { tmpW[1][kk][mn], tmpW[1][kk+1][mn] };
        v2f bII = { tmpW[2][kk][mn], tmpW[2][kk+1][mn] };
        v2f bJJ = { tmpW[3][kk][mn], tmpW[3][kk+1][mn] };
        v2f bIJ = { tmpW[4][kk][mn], tmpW[4][kk+1][mn] };
        sI  = wmma4(ah, bI,  sI);
        sJ  = wmma4(ah, bJ,  sJ);
        sII = wmma4(ah, bII, sII);
        sJJ = wmma4(ah, bJJ, sJJ);
        sIJ = wmma4(ah, bIJ, sIJ);
      }
      __syncthreads();   // tmpW + ldsIn consumed before next plane overwrites

      #pragma unroll
      for (int r = 0; r < 8; ++r) {
        const int m = r + 8*half;
        ring[0][slot][m][mn] = sI[r];
        ring[1][slot][m][mn] = sJ[r];
        ring[2][slot][m][mn] = sII[r];
        ring[3][slot][m][mn] = sJJ[r];
        ring[4][slot][m][mn] = sIJ[r];
      }
      __syncthreads();
    }

    // ---- D-direction sum (exact 7-plane sum) + fused NCC pointwise ----
    if (t >= 6) {                                 // output z = z0 + t - 6
      #pragma unroll
      for (int r = 0; r < 8; ++r) {
        const int m = r + 8*half;
        float SI = 0.f, SJ = 0.f, SII = 0.f, SJJ = 0.f, SIJ = 0.f;
        #pragma unroll
        for (int s = 0; s < 7; ++s) {
          SI  += ring[0][s][m][mn];
          SJ  += ring[1][s][m][mn];
          SII += ring[2][s][m][mn];
          SJJ += ring[3][s][m][mn];
          SIJ += ring[4][s][m][mn];
        }
        const float winsz = 343.0f;
        const float uI = SI * (1.0f / 343.0f);
        const float uJ = SJ * (1.0f / 343.0f);
        const float cross = SIJ - uJ*SI - uI*SJ + uI*uJ*winsz;
        const float Ivar  = SII - 2.0f*uI*SI + uI*uI*winsz;
        const float Jvar  = SJJ - 2.0f*uJ*SJ + uJ*uJ*winsz;
        csum += cross*cross / (Ivar*Jvar + 1e-5f);
      }
    }
  }

  // single-wave block reduction -> deterministic per-block partial
  red[lane] = csum;
  __syncthreads();
  if (lane == 0) {
    float s = 0.f;
    for (int i = 0; i < 32; ++i) s += red[i];
    partials[blockIdx.x + gridDim.x * (blockIdx.y + gridDim.y * blockIdx.z)] = s;
  }
}

__global__ void ncc_reduce(const float* __restrict__ partials, int n,
                           float* __restrict__ out) {
  if (threadIdx.x == 0 && blockIdx.x == 0) {
    double s = 0.0;
    for (int i = 0; i < n; ++i) s += (double)partials[i];
    out[0] = (float)(-s / (double)(BATCH * DD * HH * WW));
  }
}

extern "C" void kernel_launch(void* const* d_in, const int* in_sizes, int n_in,
                              void* d_out, int out_size, void* d_ws, size_t ws_size,
                              hipStream_t stream) {
  const float* I = (const float*)d_in[0];
  const float* J = (const float*)d_in[1];
  float* partials = (float*)d_ws;              // NBLK floats = 13.1 KB

  dim3 grid(WW / 16, HH / 16, BATCH * NZC);    // 14 x 12 x 20 = 3360 blocks
  ncc_main<<<grid, dim3(32, 1, 1), 0, stream>>>(I, J, partials);
  ncc_reduce<<<1, 32, 0, stream>>>(partials, NBLK, (float*)d_out);
}